// TopologicalGRFLayer_7876970021339
// MI455X (gfx1250) — compile-verified
//
#include <hip/hip_runtime.h>

#define DIM 128
#define LDS_STRIDE (DIM + 4)   // pad so 16 lanes reading stride-LDS rows hit distinct banks

typedef __attribute__((ext_vector_type(2))) float v2f;
typedef __attribute__((ext_vector_type(8))) float v8f;

// ---------------------------------------------------------------------------
// GEMM: qkv = X[M,128] @ Wqkv[128,384], split column tiles into Q / Kf / Vf.
// One wave computes one 16x16 f32 tile via V_WMMA_F32_16X16X4_F32.
// 4 waves per block share an LDS-staged 16x128 tile of X.
// A layout (16x4 f32): lanes 0-15 M=0-15 K={0,1}, lanes 16-31 M=0-15 K={2,3}.
// B layout (4x16 f32): VGPR v, lanes 0-15 -> K=v, lanes 16-31 -> K=v+2.
// ---------------------------------------------------------------------------
__global__ __launch_bounds__(128) void gemm_qkv_kernel(
    const float* __restrict__ X, const float* __restrict__ W,
    float* __restrict__ Q, float* __restrict__ Kf, float* __restrict__ Vf)
{
    __shared__ float sx[16 * LDS_STRIDE];
    const int tid  = threadIdx.x;
    const int wave = tid >> 5;
    const int lane = tid & 31;
    const int half = lane >> 4;
    const int l16  = lane & 15;
    const int m0   = blockIdx.x * 16;

    // cooperative coalesced load of 16x128 X tile into LDS
    {
        const int r  = tid >> 3;          // 0..15
        const int c0 = (tid & 7) * 16;    // 0,16,...,112
        const float4* src = (const float4*)(X + (size_t)(m0 + r) * DIM + c0);
        float4 v0 = src[0], v1 = src[1], v2 = src[2], v3 = src[3];
        float* d = sx + r * LDS_STRIDE + c0;
        *(float4*)(d + 0)  = v0;
        *(float4*)(d + 4)  = v1;
        *(float4*)(d + 8)  = v2;
        *(float4*)(d + 12) = v3;
    }
    __syncthreads();

    const int n0  = (blockIdx.y * 4 + wave) * 16;  // 0..368
    const int LDW = 3 * DIM;
    v8f c = {};
    const float* a_base = sx + l16 * LDS_STRIDE + 2 * half;
    const float* b_base = W + (size_t)(2 * half) * LDW + n0 + l16;

    #pragma unroll
    for (int k0 = 0; k0 < DIM; k0 += 4) {
        v2f a, b;
        a.x = a_base[k0];
        a.y = a_base[k0 + 1];
        b.x = b_base[(size_t)k0 * LDW];
        b.y = b_base[(size_t)(k0 + 1) * LDW];
        c = __builtin_amdgcn_wmma_f32_16x16x4_f32(
                /*neg_a=*/false, a, /*neg_b=*/false, b,
                /*c_mod=*/(short)0, c, /*reuse_a=*/false, /*reuse_b=*/false);
    }

    float* dst; int nc;
    if      (n0 <     DIM) { dst = Q;  nc = n0;           }
    else if (n0 < 2 * DIM) { dst = Kf; nc = n0 - DIM;     }
    else                   { dst = Vf; nc = n0 - 2 * DIM; }
    nc += l16;
    #pragma unroll
    for (int v = 0; v < 8; ++v)
        dst[(size_t)(m0 + v + 8 * half) * DIM + nc] = c[v];
}

// ---------------------------------------------------------------------------
__global__ void zero_kernel(float4* __restrict__ p, int n4)
{
    int i = blockIdx.x * blockDim.x + threadIdx.x;
    if (i < n4) p[i] = make_float4(0.f, 0.f, 0.f, 0.f);
}

// ---------------------------------------------------------------------------
// diffuse: Fdst[b, knn[b,n,j], :] += F[b,n,:] * inv_denom
// One block per (b,n) row; 128 threads = one float each -> each neighbor
// update is a fully coalesced burst of 128 global_atomic_add_f32 (no-return).
// ---------------------------------------------------------------------------
__global__ __launch_bounds__(128) void diffuse_kernel(
    const float* __restrict__ F, const int* __restrict__ knn,
    float* __restrict__ Fdst, int N, int K, float inv_denom)
{
    const int bn = blockIdx.x;         // b*N + n
    const int b  = bn / N;
    const int d  = threadIdx.x;
    const float v = F[(size_t)bn * DIM + d] * inv_denom;

    __shared__ int sidx[64];
    if (threadIdx.x < K) sidx[threadIdx.x] = knn[(size_t)bn * K + threadIdx.x];
    __syncthreads();

    float* base = Fdst + (size_t)b * N * DIM + d;
    for (int j = 0; j < K; ++j)
        unsafeAtomicAdd(base + (size_t)sidx[j] * DIM, v);
}

// ---------------------------------------------------------------------------
// attn[row] = dot(Q[row,:], Kf[row,:]) ; one wave32 per row
// ---------------------------------------------------------------------------
__global__ __launch_bounds__(256) void attn_kernel(
    const float* __restrict__ Q, const float* __restrict__ Kf,
    float* __restrict__ A, int M)
{
    const int row  = blockIdx.x * 8 + (threadIdx.x >> 5);
    const int lane = threadIdx.x & 31;
    if (row >= M) return;
    const float* q = Q  + (size_t)row * DIM;
    const float* k = Kf + (size_t)row * DIM;
    float s = 0.f;
    #pragma unroll
    for (int i = 0; i < DIM / 32; ++i)
        s += q[lane + 32 * i] * k[lane + 32 * i];
    #pragma unroll
    for (int off = 16; off; off >>= 1)
        s += __shfl_xor(s, off, 32);
    if (lane == 0) A[row] = s;
}

// ---------------------------------------------------------------------------
// out = (attn[:,None] * Vf) @ Wout[128,128] + bias ; WMMA f32 16x16x4.
// Scaling by attn happens while staging the A tile into LDS.
// ---------------------------------------------------------------------------
__global__ __launch_bounds__(128) void gemm_out_kernel(
    const float* __restrict__ Vf, const float* __restrict__ Attn,
    const float* __restrict__ W, const float* __restrict__ bias,
    float* __restrict__ Out)
{
    __shared__ float sx[16 * LDS_STRIDE];
    const int tid  = threadIdx.x;
    const int wave = tid >> 5;
    const int lane = tid & 31;
    const int half = lane >> 4;
    const int l16  = lane & 15;
    const int m0   = blockIdx.x * 16;

    {
        const int r  = tid >> 3;
        const int c0 = (tid & 7) * 16;
        const float scale = Attn[m0 + r];
        const float4* src = (const float4*)(Vf + (size_t)(m0 + r) * DIM + c0);
        float4 v0 = src[0], v1 = src[1], v2 = src[2], v3 = src[3];
        v0.x *= scale; v0.y *= scale; v0.z *= scale; v0.w *= scale;
        v1.x *= scale; v1.y *= scale; v1.z *= scale; v1.w *= scale;
        v2.x *= scale; v2.y *= scale; v2.z *= scale; v2.w *= scale;
        v3.x *= scale; v3.y *= scale; v3.z *= scale; v3.w *= scale;
        float* d = sx + r * LDS_STRIDE + c0;
        *(float4*)(d + 0)  = v0;
        *(float4*)(d + 4)  = v1;
        *(float4*)(d + 8)  = v2;
        *(float4*)(d + 12) = v3;
    }
    __syncthreads();

    const int n0 = (blockIdx.y * 4 + wave) * 16;  // 0..112
    v8f c = {};
    const float* a_base = sx + l16 * LDS_STRIDE + 2 * half;
    const float* b_base = W + (size_t)(2 * half) * DIM + n0 + l16;

    #pragma unroll
    for (int k0 = 0; k0 < DIM; k0 += 4) {
        v2f a, b;
        a.x = a_base[k0];
        a.y = a_base[k0 + 1];
        b.x = b_base[(size_t)k0 * DIM];
        b.y = b_base[(size_t)(k0 + 1) * DIM];
        c = __builtin_amdgcn_wmma_f32_16x16x4_f32(
                false, a, false, b, (short)0, c, false, false);
    }

    const float bv = bias[n0 + l16];
    #pragma unroll
    for (int v = 0; v < 8; ++v)
        Out[(size_t)(m0 + v + 8 * half) * DIM + n0 + l16] = c[v] + bv;
}

// ---------------------------------------------------------------------------
extern "C" void kernel_launch(void* const* d_in, const int* in_sizes, int n_in,
                              void* d_out, int out_size, void* d_ws, size_t ws_size,
                              hipStream_t stream)
{
    const float* x    = (const float*)d_in[0];
    const int*   knn  = (const int*)  d_in[1];
    const float* Wqkv = (const float*)d_in[2];
    const float* Wout = (const float*)d_in[3];
    const float* bout = (const float*)d_in[4];
    float*       out  = (float*)d_out;

    const int D = DIM;
    const int M = in_sizes[0] / D;   // B*N = 32768
    const int B = 4;
    const int N = M / B;             // 8192
    const int K = in_sizes[1] / M;   // 32
    const float inv_denom = 1.0f / ((float)K + 1e-6f);

    // workspace layout (floats): q | kfA | vfA | kfB | vfB | attn
    float* ws  = (float*)d_ws;
    const size_t seg = (size_t)M * D;
    float* q    = ws;
    float* kfA  = ws + 1 * seg;
    float* vfA  = ws + 2 * seg;
    float* kfB  = ws + 3 * seg;
    float* vfB  = ws + 4 * seg;
    float* attn = ws + 5 * seg;

    // 1) qkv projection
    gemm_qkv_kernel<<<dim3(M / 16, 6), dim3(128), 0, stream>>>(x, Wqkv, q, kfA, vfA);

    // 2) three diffusion hops (ping-pong A<->B; k/v pairs are contiguous so one
    //    zero-fill launch clears both destination buffers)
    for (int h = 0; h < 3; ++h) {
        float* ksrc = (h & 1) ? kfB : kfA;
        float* vsrc = (h & 1) ? vfB : vfA;
        float* kdst = (h & 1) ? kfA : kfB;
        float* vdst = (h & 1) ? vfA : vfB;
        const int n4 = (int)(2 * seg / 4);
        zero_kernel<<<(n4 + 255) / 256, 256, 0, stream>>>((float4*)kdst, n4);
        diffuse_kernel<<<M, 128, 0, stream>>>(vsrc, knn, vdst, N, K, inv_denom);
        diffuse_kernel<<<M, 128, 0, stream>>>(ksrc, knn, kdst, N, K, inv_denom);
    }
    // after 3 hops results live in the B buffers

    // 3) attention scalar per row
    attn_kernel<<<M / 8, 256, 0, stream>>>(q, kfB, attn, M);

    // 4) output projection with bias
    gemm_out_kernel<<<dim3(M / 16, 2), dim3(128), 0, stream>>>(vfB, attn, Wout, bout, out);
}